// IADDPlugin_44693429682554
// MI455X (gfx1250) — compile-verified
//
#include <hip/hip_runtime.h>
#include <math.h>

typedef __attribute__((ext_vector_type(2))) float v2f;
typedef __attribute__((ext_vector_type(8))) float v8f;

#define B_ROWS 4096
#define C_COLS 8192
#define D_FEAT 512
#define TOPK   10
#define T_TEMP 2.0f
#define EPS_D  1e-12f

// ---------------------------------------------------------------------------
// 1) L2-normalize feature rows; also emit per-row squared norm of the
//    normalized row (used by the euclidean-distance formula).
// ---------------------------------------------------------------------------
__global__ void __launch_bounds__(256) l2norm_kernel(const float* __restrict__ f,
                                                     float* __restrict__ fn,
                                                     float* __restrict__ xx) {
  __shared__ float red[256];
  const int i = blockIdx.x;
  const int t = threadIdx.x;
  const float* r = f + (size_t)i * D_FEAT;
  float s = 0.f;
  for (int j = t; j < D_FEAT; j += 256) { const float v = r[j]; s += v * v; }
  red[t] = s; __syncthreads();
  for (int off = 128; off > 0; off >>= 1) {
    if (t < off) red[t] += red[t + off];
    __syncthreads();
  }
  const float ss  = red[0];
  const float inv = 1.f / fmaxf(sqrtf(ss), 1e-12f);
  for (int j = t; j < D_FEAT; j += 256) fn[(size_t)i * D_FEAT + j] = r[j] * inv;
  if (t == 0) xx[i] = ss * inv * inv;
}

// ---------------------------------------------------------------------------
// 2) sim = F * F^T  via V_WMMA_F32_16X16X4_F32 (fp32 matrix pipe).
//    Block = 256 thr (8 wave32) -> 128x128 tile; LDS-staged 32-wide K chunks.
//    Fragment layouts per CDNA5 ISA 7.12.2:
//      A (16x4 f32, 2 VGPR): lane L -> row M=L%16; VGPR pair holds
//        K = 2*(L/16) and K = 2*(L/16)+1.
//      B (4x16, 2 VGPR): lane L -> col N=L%16; same K pairing.
//        For F*F^T this equals the A-load of the column block.
//      C/D (16x16 f32, 8 VGPR): elem i -> M = i + 8*(L/16), N = L%16.
// ---------------------------------------------------------------------------
__global__ void __launch_bounds__(256) simgemm_kernel(const float* __restrict__ F,
                                                      float* __restrict__ out) {
  __shared__ float As[128 * 32];
  __shared__ float Bs[128 * 32];
  const int tid  = threadIdx.x;
  const int lane = tid & 31;
  const int wave = tid >> 5;
  const int half = lane >> 4;
  const int lm   = lane & 15;
  const int m0   = blockIdx.y * 128;
  const int n0   = blockIdx.x * 128;
  v8f acc[8] = {};
  for (int k0 = 0; k0 < D_FEAT; k0 += 32) {
#pragma unroll
    for (int q = 0; q < 4; ++q) {
      const int seg = tid + 256 * q;          // 1024 float4 segments per buffer
      const int r   = seg >> 3;
      const int kq  = (seg & 7) << 2;
      *(float4*)&As[r * 32 + kq] = *(const float4*)&F[(size_t)(m0 + r) * D_FEAT + k0 + kq];
      *(float4*)&Bs[r * 32 + kq] = *(const float4*)&F[(size_t)(n0 + r) * D_FEAT + k0 + kq];
    }
    if (k0 + 32 < D_FEAT)  // pull next K-chunk toward the caches (global_prefetch_b8)
      __builtin_prefetch(&F[(size_t)(m0 + (tid >> 1)) * D_FEAT + k0 + 32 + (tid & 1) * 16], 0, 1);
    __syncthreads();
#pragma unroll
    for (int ks = 0; ks < 8; ++ks) {
      const int kk = 4 * ks + 2 * half;
      const v2f a = *(const v2f*)&As[(16 * wave + lm) * 32 + kk];
#pragma unroll
      for (int nt = 0; nt < 8; ++nt) {
        const v2f b = *(const v2f*)&Bs[(16 * nt + lm) * 32 + kk];
        acc[nt] = __builtin_amdgcn_wmma_f32_16x16x4_f32(
            /*neg_a=*/false, a, /*neg_b=*/false, b,
            /*c_mod=*/(short)0, acc[nt], /*reuse_a=*/false, /*reuse_b=*/false);
      }
    }
    __syncthreads();
  }
#pragma unroll
  for (int nt = 0; nt < 8; ++nt)
#pragma unroll
    for (int i = 0; i < 8; ++i)
      out[(size_t)(m0 + 16 * wave + i + 8 * half) * B_ROWS + (n0 + 16 * nt + lm)] = acc[nt][i];
}

// ---------------------------------------------------------------------------
// 3) MCD per row: mean of positives (same label, off-diagonal) minus mean of
//    top-10 "negatives" (note: reference's is_neg INCLUDES the diagonal).
//    Per-thread insertion top-10 + deterministic LDS tree merge.
// ---------------------------------------------------------------------------
__global__ void __launch_bounds__(256) mcd_kernel(const float* __restrict__ sim,
                                                  const int* __restrict__ labels,
                                                  float* __restrict__ mcd) {
  __shared__ float tk[256][TOPK];
  __shared__ float sps[256];
  __shared__ int   spc[256];
  const int i  = blockIdx.x;
  const int t  = threadIdx.x;
  const int li = labels[i];
  const float* row = sim + (size_t)i * B_ROWS;
  float top[TOPK];
#pragma unroll
  for (int k = 0; k < TOPK; ++k) top[k] = -__builtin_huge_valf();
  float psum = 0.f; int pcnt = 0;
  for (int j = t; j < B_ROWS; j += 256) {
    const float v = row[j];
    if ((labels[j] == li) && (j != i)) {
      psum += v; ++pcnt;
    } else if (v > top[TOPK - 1]) {
      int k = TOPK - 1;
      while (k > 0 && top[k - 1] < v) { top[k] = top[k - 1]; --k; }
      top[k] = v;
    }
  }
  sps[t] = psum; spc[t] = pcnt;
#pragma unroll
  for (int k = 0; k < TOPK; ++k) tk[t][k] = top[k];
  __syncthreads();
  for (int off = 128; off > 0; off >>= 1) {
    if (t < off) {
      sps[t] += sps[t + off];
      spc[t] += spc[t + off];
      float m[TOPK];
      int ia = 0, ib = 0;
#pragma unroll
      for (int k = 0; k < TOPK; ++k) {
        const float va = tk[t][ia], vb = tk[t + off][ib];
        if (va >= vb) { m[k] = va; ++ia; } else { m[k] = vb; ++ib; }
      }
#pragma unroll
      for (int k = 0; k < TOPK; ++k) tk[t][k] = m[k];
    }
    __syncthreads();
  }
  if (t == 0) {
    const float ps = (spc[0] > 0) ? (sps[0] / (float)spc[0]) : 0.f;
    float ns = 0.f;
#pragma unroll
    for (int k = 0; k < TOPK; ++k) ns += tk[0][k];
    mcd[i] = ps - ns * (1.f / (float)TOPK);
  }
}

// ---------------------------------------------------------------------------
// 4) alpha = sigmoid((mcd1 - mcd2) * T)
// ---------------------------------------------------------------------------
__global__ void alpha_kernel(const float* __restrict__ mcd1,
                             const float* __restrict__ mcd2,
                             float* __restrict__ alpha) {
  const int i = blockIdx.x * blockDim.x + threadIdx.x;
  if (i < B_ROWS) {
    const float d = (mcd1[i] - mcd2[i]) * T_TEMP;
    alpha[i] = 1.f / (1.f + expf(-d));
  }
}

// ---------------------------------------------------------------------------
// 5) Per-row mutual-distillation loss. Both 8192-wide logits rows are staged
//    once in LDS (64 KB/WG; CDNA5 WGP has 320 KB) so HBM sees each element once.
//    sum p1*(lp1 - lp2) = U12/S1 - ((m1+logS1) - (m2+logS2)), U12 = sum e1*(x1-x2).
// ---------------------------------------------------------------------------
__global__ void __launch_bounds__(256) rowloss_kernel(const float* __restrict__ x1,
                                                      const float* __restrict__ x2,
                                                      const float* __restrict__ alpha,
                                                      float* __restrict__ rowloss) {
  __shared__ float L1[C_COLS];
  __shared__ float L2[C_COLS];
  __shared__ float red[256];
  __shared__ float bcast[6];  // M1, M2, S1, S2, U12, U21
  const int i = blockIdx.x;
  const int t = threadIdx.x;
  const float* r1 = x1 + (size_t)i * C_COLS;
  const float* r2 = x2 + (size_t)i * C_COLS;
  for (int j = t; j < C_COLS; j += 256) { L1[j] = r1[j]; L2[j] = r2[j]; }
  __syncthreads();

  float m1 = -__builtin_huge_valf(), m2 = -__builtin_huge_valf();
  for (int j = t; j < C_COLS; j += 256) { m1 = fmaxf(m1, L1[j]); m2 = fmaxf(m2, L2[j]); }
  red[t] = m1; __syncthreads();
  for (int off = 128; off > 0; off >>= 1) { if (t < off) red[t] = fmaxf(red[t], red[t + off]); __syncthreads(); }
  if (t == 0) bcast[0] = red[0];
  __syncthreads();
  red[t] = m2; __syncthreads();
  for (int off = 128; off > 0; off >>= 1) { if (t < off) red[t] = fmaxf(red[t], red[t + off]); __syncthreads(); }
  if (t == 0) bcast[1] = red[0];
  __syncthreads();

  const float M1 = bcast[0], M2 = bcast[1];
  float S1 = 0.f, S2 = 0.f, U12 = 0.f, U21 = 0.f;
  for (int j = t; j < C_COLS; j += 256) {
    const float a = L1[j], b = L2[j];
    const float e1 = expf(a - M1);
    const float e2 = expf(b - M2);
    S1 += e1; S2 += e2;
    U12 += e1 * (a - b);
    U21 += e2 * (b - a);
  }
  float vals[4] = {S1, S2, U12, U21};
  for (int v = 0; v < 4; ++v) {
    red[t] = vals[v]; __syncthreads();
    for (int off = 128; off > 0; off >>= 1) { if (t < off) red[t] += red[t + off]; __syncthreads(); }
    if (t == 0) bcast[2 + v] = red[0];
    __syncthreads();
  }
  if (t == 0) {
    const float A1 = M1 + logf(bcast[2]);
    const float A2 = M2 + logf(bcast[3]);
    const float loss1 = bcast[4] / bcast[2] - (A1 - A2);  // m1 teaches m2
    const float loss2 = bcast[5] / bcast[3] - (A2 - A1);  // m2 teaches m1
    const float al = alpha[i];
    rowloss[i] = al * loss1 + (1.f - al) * loss2;
  }
}

// ---------------------------------------------------------------------------
// 6) hybrid_dist: h holds sim2 (stored straight into d_out's hybrid slot by
//    the second GEMM) and is overwritten elementwise (1 read + 1 write each).
// ---------------------------------------------------------------------------
__global__ void __launch_bounds__(256) hybrid_kernel(const float* __restrict__ sim1,
                                                     float* __restrict__ h,
                                                     const float* __restrict__ alpha,
                                                     const float* __restrict__ xx1,
                                                     const float* __restrict__ xx2) {
  const size_t p = (size_t)blockIdx.x * 256 + threadIdx.x;
  const int i = (int)(p >> 12);
  const int j = (int)(p & (B_ROWS - 1));
  const float a  = alpha[i];
  const float s1 = sim1[p];
  const float s2 = h[p];
  const float d1 = sqrtf(fmaxf(xx1[i] + xx1[j] - 2.f * s1, EPS_D));
  const float d2 = sqrtf(fmaxf(xx2[i] + xx2[j] - 2.f * s2, EPS_D));
  h[p] = a * d1 + (1.f - a) * d2;
}

// ---------------------------------------------------------------------------
// 7) Deterministic single-block means for the four scalar outputs.
// ---------------------------------------------------------------------------
__global__ void __launch_bounds__(256) finalize_kernel(const float* __restrict__ rowloss,
                                                       const float* __restrict__ mcd1,
                                                       const float* __restrict__ mcd2,
                                                       const float* __restrict__ alpha,
                                                       float* __restrict__ out) {
  __shared__ float red[256];
  __shared__ float res[4];
  const int t = threadIdx.x;
  const float* srcs[4] = {rowloss, mcd1, mcd2, alpha};
  for (int v = 0; v < 4; ++v) {
    float s = 0.f;
    for (int j = t; j < B_ROWS; j += 256) s += srcs[v][j];
    red[t] = s; __syncthreads();
    for (int off = 128; off > 0; off >>= 1) { if (t < off) red[t] += red[t + off]; __syncthreads(); }
    if (t == 0) res[v] = red[0] * (1.f / (float)B_ROWS);
    __syncthreads();
  }
  if (t == 0) {
    const size_t bb = (size_t)B_ROWS * B_ROWS;
    out[0]      = res[0] * 1.0f;  // LAMBDA_DISTILL = 1.0
    out[1 + bb] = res[1];         // mcd_m1.mean()
    out[2 + bb] = res[2];         // mcd_m2.mean()
    out[3 + bb] = res[3];         // alpha.mean()
  }
}

// ---------------------------------------------------------------------------
extern "C" void kernel_launch(void* const* d_in, const int* in_sizes, int n_in,
                              void* d_out, int out_size, void* d_ws, size_t ws_size,
                              hipStream_t stream) {
  const float* logits1 = (const float*)d_in[0];
  const float* logits2 = (const float*)d_in[1];
  const float* feats1  = (const float*)d_in[2];
  const float* feats2  = (const float*)d_in[3];
  const int*   labels  = (const int*)d_in[4];
  float* out = (float*)d_out;

  // workspace layout (floats): f1n, f2n, sim1, xx1, xx2, mcd1, mcd2, alpha, rloss
  float* ws    = (float*)d_ws;
  float* f1n   = ws;
  float* f2n   = f1n  + (size_t)B_ROWS * D_FEAT;
  float* sim1  = f2n  + (size_t)B_ROWS * D_FEAT;
  float* xx1   = sim1 + (size_t)B_ROWS * B_ROWS;
  float* xx2   = xx1  + B_ROWS;
  float* mcd1  = xx2  + B_ROWS;
  float* mcd2  = mcd1 + B_ROWS;
  float* alpha = mcd2 + B_ROWS;
  float* rloss = alpha + B_ROWS;
  float* sim2  = out + 1;  // hybrid-dist output slot doubles as sim2 storage

  l2norm_kernel<<<B_ROWS, 256, 0, stream>>>(feats1, f1n, xx1);
  l2norm_kernel<<<B_ROWS, 256, 0, stream>>>(feats2, f2n, xx2);

  dim3 gg(B_ROWS / 128, B_ROWS / 128);
  simgemm_kernel<<<gg, 256, 0, stream>>>(f1n, sim1);
  simgemm_kernel<<<gg, 256, 0, stream>>>(f2n, sim2);

  mcd_kernel<<<B_ROWS, 256, 0, stream>>>(sim1, labels, mcd1);
  mcd_kernel<<<B_ROWS, 256, 0, stream>>>(sim2, labels, mcd2);

  alpha_kernel<<<B_ROWS / 256, 256, 0, stream>>>(mcd1, mcd2, alpha);

  rowloss_kernel<<<B_ROWS, 256, 0, stream>>>(logits1, logits2, alpha, rloss);

  const unsigned hybrid_blocks = (unsigned)(((size_t)B_ROWS * B_ROWS) / 256);
  hybrid_kernel<<<hybrid_blocks, 256, 0, stream>>>(sim1, sim2, alpha, xx1, xx2);

  finalize_kernel<<<1, 256, 0, stream>>>(rloss, mcd1, mcd2, alpha, out);
}